// DLRLoss_13967233647263
// MI455X (gfx1250) — compile-verified
//
#include <hip/hip_runtime.h>
#include <hip/hip_bf16.h>
#include <stdint.h>

// ---------------- configuration (problem: B=65536 rows, C=1000 cols f32) ----
#define WAVE      32
#define BLOCK     256
#define WPB       (BLOCK / WAVE)   // 8 waves per block
#define NBLK      2048             // 16384 waves total -> 4 rows per wave
#define BUF_F     1024             // LDS floats per buffer (>= C, covers 4096B)

// ---------------- LDS address helper (AS3 byte offset for inline asm) -------
typedef __attribute__((address_space(3))) void lds_void_t;

__device__ __forceinline__ unsigned to_lds_off(void* p) {
  return (unsigned)(unsigned long long)(lds_void_t*)p;
}

// One async B128 issue: IOFFSET applies to BOTH the global source address and
// the LDS destination (LDS address = VDST VGPR + offset), so all
// full-row issues share one VGPR address pair + one LDS-address VGPR.
#define ASYNC_LD_OFF(la, ga, OFFS)                                             \
  asm volatile("global_load_async_to_lds_b128 %0, %1, off offset:" #OFFS       \
               " th:TH_LOAD_NT" :: "v"(la), "v"(ga) : "memory")

// ---------------- async row copy: global -> LDS via gfx1250 async path ------
// Copies one row (C floats, C*4 <= 4096 bytes, row base 16B aligned) into the
// wave's LDS buffer using GLOBAL_LOAD_ASYNC_TO_LDS_B128 (tracked by ASYNCcnt).
// TH_LOAD_NT: the 262MB input is streamed exactly once (> 192MB L2), so
// non-temporal avoids L2 pollution on the way through. 8 issues per row.
__device__ __forceinline__ void async_copy_row(const float* __restrict__ rowbase,
                                               unsigned lds_base, int lane, int row_bytes) {
  const unsigned lane_b = (unsigned)(lane * 16);
  const unsigned long long ga = (unsigned long long)(const char*)rowbase + lane_b;
  const unsigned           la = lds_base + lane_b;

  // issues 0..6: max byte = 3072 + 31*16 + 15 = 3583 < 4000, always in-row
  ASYNC_LD_OFF(la, ga, 0);
  ASYNC_LD_OFF(la, ga, 512);
  ASYNC_LD_OFF(la, ga, 1024);
  ASYNC_LD_OFF(la, ga, 1536);
  ASYNC_LD_OFF(la, ga, 2048);
  ASYNC_LD_OFF(la, ga, 2560);
  ASYNC_LD_OFF(la, ga, 3072);

  // issue 7 (tail): lanes past row end clamp to row base (stay in-bounds);
  // their LDS slots (>= row_bytes) are scratch the compute never reads.
  const int off7  = 3584 + (int)lane_b;
  const int soff7 = (off7 < row_bytes) ? off7 : 0;
  const unsigned long long ga7 = (unsigned long long)(const char*)rowbase + (unsigned)soff7;
  const unsigned           la7 = lds_base + (unsigned)off7;
  asm volatile("global_load_async_to_lds_b128 %0, %1, off th:TH_LOAD_NT"
               :: "v"(la7), "v"(ga7) : "memory");
}

// ---------------- kernel 1: per-row DLR loss, per-wave partial sums ---------
__global__ void __launch_bounds__(BLOCK)
dlr_rowloss_kernel(const float* __restrict__ x,
                   const long long* __restrict__ tgt,
                   float* __restrict__ partials,
                   int B, int C, int NW, int R, int row_bytes) {
  __shared__ __align__(16) float smem[WPB][2][BUF_F];

  const int lane   = threadIdx.x & (WAVE - 1);
  const int wlocal = threadIdx.x >> 5;
  const int w      = blockIdx.x * WPB + wlocal;

  float* buf0 = &smem[wlocal][0][0];
  float* buf1 = &smem[wlocal][1][0];
  const unsigned lb0 = to_lds_off(buf0);
  const unsigned lb1 = to_lds_off(buf1);

  float wsum = 0.0f;

  // prologue: kick off the first row's async copy
  if (w < B) async_copy_row(x + (size_t)w * C, lb0, lane, row_bytes);

  for (int i = 0; i < R; ++i) {
    const int row = w + i * NW;
    if (row >= B) break;

    const int  nrow      = row + NW;
    const bool have_next = (i + 1 < R) && (nrow < B);
    if (have_next)
      async_copy_row(x + (size_t)nrow * C, ((i + 1) & 1) ? lb1 : lb0, lane, row_bytes);

    // wait for current row's 8 async loads (next row's 8 may stay in flight)
    if (have_next) asm volatile("s_wait_asynccnt 0x8" ::: "memory");
    else           asm volatile("s_wait_asynccnt 0x0" ::: "memory");

    const float* buf = (i & 1) ? buf1 : buf0;
    // wave-uniform target column -> SGPR
    const int t = __builtin_amdgcn_readfirstlane((int)tgt[row]);

    // per-lane top-3 (a1>=a2>=a3), argmax index (lowest on ties), x_y pickup
    float a1 = -INFINITY, a2 = -INFINITY, a3 = -INFINITY;
    int   i1 = 0x7fffffff;
    float xy = 0.0f;

#pragma unroll
    for (int c = 0; c < (BUF_F / 128); ++c) {
      const int base = c * 128 + lane * 4;
      if (base < C) {
        const float4 v4 = *reinterpret_cast<const float4*>(buf + base);
        const float vv[4] = {v4.x, v4.y, v4.z, v4.w};
#pragma unroll
        for (int k = 0; k < 4; ++k) {
          const int   j = base + k;
          const float v = vv[k];
          if (j == t) xy = v;
          if (v > a1)      { a3 = a2; a2 = a1; a1 = v; i1 = j; }
          else if (v > a2) { a3 = a2; a2 = v; }
          else if (v > a3) { a3 = v; }
        }
      }
    }

    // wave32 butterfly merge of (top3, argmax, xy)
#pragma unroll
    for (int off = 16; off > 0; off >>= 1) {
      const float b1 = __shfl_xor(a1, off, WAVE);
      const float b2 = __shfl_xor(a2, off, WAVE);
      const float b3 = __shfl_xor(a3, off, WAVE);
      const int   bi = __shfl_xor(i1, off, WAVE);
      xy            += __shfl_xor(xy, off, WAVE);

      if (b1 > a1 || (b1 == a1 && bi < i1)) i1 = bi;   // lowest-index argmax

      float r1, r2, r3;
      if (a1 >= b1) {
        r1 = a1;
        if (a2 >= b1) { r2 = a2; r3 = fmaxf(a3, b1); }
        else          { r2 = b1; r3 = fmaxf(a2, b2); }
      } else {
        r1 = b1;
        if (b2 >= a1) { r2 = b2; r3 = fmaxf(b3, a1); }
        else          { r2 = a1; r3 = fmaxf(a2, b2); }
      }
      a1 = r1; a2 = r2; a3 = r3;
    }

    // loss_indiv = -(x_y - z2*ind - z1*(1-ind)) / (z1 - z3 + eps)
    const float ztop = (i1 == t) ? a2 : a1;
    wsum += -(xy - ztop) / (a1 - a3 + 1e-12f);
  }

  if (lane == 0) partials[w] = wsum;
}

// ---------------- kernel 2: deterministic reduce of wave partials ----------
__global__ void __launch_bounds__(BLOCK)
dlr_reduce_kernel(const float* __restrict__ partials, int n,
                  float* __restrict__ out, float inv_b) {
  __shared__ float s[BLOCK];
  float acc = 0.0f;
  for (int i = threadIdx.x; i < n; i += BLOCK) acc += partials[i];  // fixed order
  s[threadIdx.x] = acc;
  __syncthreads();
#pragma unroll
  for (int stride = BLOCK / 2; stride > 0; stride >>= 1) {
    if ((int)threadIdx.x < stride) s[threadIdx.x] += s[threadIdx.x + stride];
    __syncthreads();
  }
  if (threadIdx.x == 0) out[0] = s[0] * inv_b;
}

// ---------------- entry point ----------------------------------------------
extern "C" void kernel_launch(void* const* d_in, const int* in_sizes, int n_in,
                              void* d_out, int out_size, void* d_ws, size_t ws_size,
                              hipStream_t stream) {
  const float*     x   = (const float*)d_in[0];
  const long long* tgt = (const long long*)d_in[1];
  const int B = in_sizes[1];             // 65536
  const int C = in_sizes[0] / B;         // 1000

  const int NW = NBLK * WPB;             // 16384 waves
  const int R  = (B + NW - 1) / NW;      // 4 rows per wave
  const int row_bytes = C * 4;           // 4000

  float* partials = (float*)d_ws;        // NW floats = 64 KB scratch

  dlr_rowloss_kernel<<<dim3(NBLK), dim3(BLOCK), 0, stream>>>(x, tgt, partials,
                                                             B, C, NW, R, row_bytes);
  dlr_reduce_kernel<<<dim3(1), dim3(BLOCK), 0, stream>>>(partials, NW,
                                                         (float*)d_out, 1.0f / (float)B);
}